// GNNModel_29429115912637
// MI455X (gfx1250) — compile-verified
//
#include <hip/hip_runtime.h>
#include <hip/hip_bf16.h>
#include <math.h>

// ---------------------------------------------------------------------------
// GCN on gfx1250 (MI455X, wave32): CSR-gather aggregation + f32 WMMA GEMMs.
// ---------------------------------------------------------------------------

typedef __attribute__((ext_vector_type(2))) float v2f;
typedef __attribute__((ext_vector_type(8))) float v8f;

// ---------------------------- CSR construction -----------------------------

__global__ void init_cnt_kernel(int* __restrict__ cnt, int n) {
    int i = blockIdx.x * blockDim.x + threadIdx.x;
    if (i < n) cnt[i] = 1;  // self-loop contributes 1 to every node's degree
}

__global__ void count_edges_kernel(const int* __restrict__ src,
                                   const int* __restrict__ dst,
                                   int* __restrict__ cnt, int ne) {
    int e = blockIdx.x * blockDim.x + threadIdx.x;
    if (e < ne) {
        int s = src[e], d = dst[e];
        if (s != d) atomicAdd(&cnt[d], 1);  // existing self-loops have ew=0 -> skip
    }
}

#define SCAN_T 1024
__global__ void scan_kernel(const int* __restrict__ cnt,
                            int* __restrict__ row_ptr,
                            int* __restrict__ fill_cnt,
                            float* __restrict__ dis, int n) {
    __shared__ int part[SCAN_T];
    int t = threadIdx.x;
    int per = (n + SCAN_T - 1) / SCAN_T;
    int beg = t * per;
    int end = beg + per; if (end > n) end = n;
    int s = 0;
    for (int i = beg; i < end; ++i) s += cnt[i];
    part[t] = s;
    __syncthreads();
    // Hillis-Steele inclusive scan over the 1024 partials
    for (int off = 1; off < SCAN_T; off <<= 1) {
        int v = (t >= off) ? part[t - off] : 0;
        __syncthreads();
        part[t] += v;
        __syncthreads();
    }
    int run = (t == 0) ? 0 : part[t - 1];   // exclusive prefix for this chunk
    for (int i = beg; i < end; ++i) {
        row_ptr[i]  = run;
        fill_cnt[i] = run;
        dis[i] = rsqrtf((float)cnt[i]);     // deg >= 1 always (self-loop)
        run += cnt[i];
    }
    if (t == SCAN_T - 1) row_ptr[n] = part[SCAN_T - 1];
}

__global__ void fill_edges_kernel(const int* __restrict__ src,
                                  const int* __restrict__ dst,
                                  const float* __restrict__ dis,
                                  int* __restrict__ fill_cnt,
                                  int* __restrict__ col_idx,
                                  float* __restrict__ col_w,
                                  int ne, int n) {
    int e = blockIdx.x * blockDim.x + threadIdx.x;
    if (e < ne) {
        int s = src[e], d = dst[e];
        if (s != d) {
            int pos = atomicAdd(&fill_cnt[d], 1);
            col_idx[pos] = s;
            col_w[pos]   = dis[s] * dis[d];
        }
    } else if (e < ne + n) {
        int i = e - ne;                      // added self-loop, weight 1
        int pos = atomicAdd(&fill_cnt[i], 1);
        col_idx[pos] = i;
        col_w[pos]   = dis[i] * dis[i];
    }
}

// ------------------------------ WMMA GEMM ----------------------------------
// C[M x Nc] = A[M x K] * B[K x Nc] (+ bias). M % 16 == 0, K % 4 == 0,
// Nc % 64 == 0. Each wave computes a 16x64 strip: 4 accumulators, A-fragment
// reused across the 4 column tiles. Uses V_WMMA_F32_16X16X4_F32 (full f32).
//
// A 16x4 fragment layout (ISA 7.12.2, 32-bit A): lanes 0-15 hold M=lane,
//   VGPR0 = K+0 (half 0) / K+2 (half 1), VGPR1 = K+1 / K+3.
// B 4x16 fragment: row (K) striped across lanes within a VGPR, same K split.
// C 16x16: VGPR r -> M = r + 8*half, N = lane & 15.

__global__ void gemm_f32_wmma(const float* __restrict__ A,
                              const float* __restrict__ B,
                              const float* __restrict__ bias,
                              float* __restrict__ C,
                              int M, int K, int Nc) {
    const int lane = threadIdx.x & 31;
    const int wave = threadIdx.x >> 5;
    const int tiles_n = Nc >> 6;                       // 64 cols per wave
    const int tile = blockIdx.x * (blockDim.x >> 5) + wave;
    const int total = (M >> 4) * tiles_n;
    if (tile >= total) return;
    const int tm = tile / tiles_n;
    const int tn = tile - tm * tiles_n;
    const int row0 = tm << 4;
    const int col0 = tn << 6;

    const int half = lane >> 4;                        // 0 or 1
    const int l15  = lane & 15;

    v8f c0 = {}, c1 = {}, c2 = {}, c3 = {};
    const float* Ap = A + (size_t)(row0 + l15) * K + 2 * half;
    const float* Bp = B + (size_t)(2 * half) * Nc + col0 + l15;

    for (int k = 0; k < K; k += 4) {
        v2f a;
        a.x = Ap[k];
        a.y = Ap[k + 1];
        const float* bk = Bp + (size_t)k * Nc;
        v2f b0, b1, b2, b3;
        b0.x = bk[0];       b0.y = bk[Nc];
        b1.x = bk[16];      b1.y = bk[Nc + 16];
        b2.x = bk[32];      b2.y = bk[Nc + 32];
        b3.x = bk[48];      b3.y = bk[Nc + 48];
        c0 = __builtin_amdgcn_wmma_f32_16x16x4_f32(false, a, false, b0, (short)0, c0, false, false);
        c1 = __builtin_amdgcn_wmma_f32_16x16x4_f32(false, a, false, b1, (short)0, c1, false, false);
        c2 = __builtin_amdgcn_wmma_f32_16x16x4_f32(false, a, false, b2, (short)0, c2, false, false);
        c3 = __builtin_amdgcn_wmma_f32_16x16x4_f32(false, a, false, b3, (short)0, c3, false, false);
    }

    float bb0 = 0.f, bb1 = 0.f, bb2 = 0.f, bb3 = 0.f;
    if (bias) {
        bb0 = bias[col0 + l15];
        bb1 = bias[col0 + l15 + 16];
        bb2 = bias[col0 + l15 + 32];
        bb3 = bias[col0 + l15 + 48];
    }
    #pragma unroll
    for (int r = 0; r < 8; ++r) {
        int row = row0 + r + 8 * half;
        float* out = C + (size_t)row * Nc + col0 + l15;
        out[0]  = c0[r] + bb0;
        out[16] = c1[r] + bb1;
        out[32] = c2[r] + bb2;
        out[48] = c3[r] + bb3;
    }
}

// --------------------------- CSR gather-aggregate --------------------------
// out[v,:] = sum_{j in row v} h[col_idx[j],:] * col_w[j] + bias (opt. ReLU).
// One block per node, float2 per thread (256 threads * 2 = D = 512).

__global__ void aggregate_kernel(const float* __restrict__ h,
                                 const int* __restrict__ row_ptr,
                                 const int* __restrict__ col_idx,
                                 const float* __restrict__ col_w,
                                 const float* __restrict__ bias,
                                 float* __restrict__ out,
                                 int D, int do_relu) {
    const int node = blockIdx.x;
    const int f = threadIdx.x * 2;
    if (f >= D) return;
    const int beg = row_ptr[node];
    const int end = row_ptr[node + 1];
    float ax = 0.f, ay = 0.f;
    for (int j = beg; j < end; ++j) {
        const int s   = col_idx[j];
        const float w = col_w[j];
        const float2 hv = *(const float2*)(h + (size_t)s * D + f);
        ax = fmaf(hv.x, w, ax);
        ay = fmaf(hv.y, w, ay);
    }
    ax += bias[f];
    ay += bias[f + 1];
    if (do_relu) { ax = fmaxf(ax, 0.f); ay = fmaxf(ay, 0.f); }
    float2 o; o.x = ax; o.y = ay;
    *(float2*)(out + (size_t)node * D + f) = o;
}

// ------------------------------ log_softmax --------------------------------
// One wave32 per node, 2 classes per lane (C = 64).

__global__ void log_softmax_kernel(const float* __restrict__ logits,
                                   float* __restrict__ lsm, int n, int C) {
    const int gwave = (blockIdx.x * blockDim.x + threadIdx.x) >> 5;
    const int lane = threadIdx.x & 31;
    if (gwave >= n) return;
    const float* row = logits + (size_t)gwave * C;
    float2 v = *(const float2*)(row + lane * 2);
    float m = fmaxf(v.x, v.y);
    #pragma unroll
    for (int off = 16; off > 0; off >>= 1) m = fmaxf(m, __shfl_xor(m, off, 32));
    float s = __expf(v.x - m) + __expf(v.y - m);
    #pragma unroll
    for (int off = 16; off > 0; off >>= 1) s += __shfl_xor(s, off, 32);
    const float lse = m + __logf(s);
    float2 o; o.x = v.x - lse; o.y = v.y - lse;
    *(float2*)(lsm + (size_t)gwave * C + lane * 2) = o;
}

// ------------------------------- launcher ----------------------------------

extern "C" void kernel_launch(void* const* d_in, const int* in_sizes, int n_in,
                              void* d_out, int out_size, void* d_ws, size_t ws_size,
                              hipStream_t stream) {
    (void)n_in; (void)out_size; (void)ws_size;

    const float* x  = (const float*)d_in[0];
    const int*   ei = (const int*)d_in[1];     // edge_index [2, E]
    const float* W1 = (const float*)d_in[2];
    const float* b1 = (const float*)d_in[3];
    const float* W2 = (const float*)d_in[4];
    const float* b2 = (const float*)d_in[5];
    const float* Wc = (const float*)d_in[6];
    const float* bc = (const float*)d_in[7];

    const int D  = 512;
    const int C  = 64;
    const int N  = in_sizes[0] / D;            // 10000
    const int E  = in_sizes[1] / 2;            // 160000
    const int* src = ei;
    const int* dst = ei + E;

    // ---- carve workspace (256B aligned) ----
    char* w = (char*)d_ws;
    auto carve = [&](size_t bytes) -> char* {
        char* p = w;
        w += (bytes + 255) & ~(size_t)255;
        return p;
    };
    int*   cnt      = (int*)  carve((size_t)N * 4);
    int*   row_ptr  = (int*)  carve((size_t)(N + 1) * 4);
    int*   fill_cnt = (int*)  carve((size_t)N * 4);
    float* dis      = (float*)carve((size_t)N * 4);
    int*   col_idx  = (int*)  carve((size_t)(E + N) * 4);
    float* col_w    = (float*)carve((size_t)(E + N) * 4);
    float* bufA     = (float*)carve((size_t)N * D * 4);
    float* bufB     = (float*)carve((size_t)N * D * 4);

    float* logits = (float*)d_out;             // [N, C]
    float* lsm    = logits + (size_t)N * C;    // [N, C]

    // ---- graph normalization: degrees -> scan -> CSR fill ----
    init_cnt_kernel<<<(N + 255) / 256, 256, 0, stream>>>(cnt, N);
    count_edges_kernel<<<(E + 255) / 256, 256, 0, stream>>>(src, dst, cnt, E);
    scan_kernel<<<1, SCAN_T, 0, stream>>>(cnt, row_ptr, fill_cnt, dis, N);
    fill_edges_kernel<<<(E + N + 255) / 256, 256, 0, stream>>>(
        src, dst, dis, fill_cnt, col_idx, col_w, E, N);

    // ---- layer 1: h = x @ W1 ; agg + b1 + ReLU ----
    {
        int tiles = (N / 16) * (D / 64);
        gemm_f32_wmma<<<(tiles + 7) / 8, 256, 0, stream>>>(x, W1, nullptr, bufA, N, D, D);
    }
    aggregate_kernel<<<N, 256, 0, stream>>>(bufA, row_ptr, col_idx, col_w, b1, bufB, D, 1);

    // ---- layer 2: h = a1 @ W2 ; agg + b2 ----
    {
        int tiles = (N / 16) * (D / 64);
        gemm_f32_wmma<<<(tiles + 7) / 8, 256, 0, stream>>>(bufB, W2, nullptr, bufA, N, D, D);
    }
    aggregate_kernel<<<N, 256, 0, stream>>>(bufA, row_ptr, col_idx, col_w, b2, bufB, D, 0);

    // ---- classifier: logits = a2 @ Wc + bc ----
    {
        int tiles = (N / 16) * (C / 64);
        gemm_f32_wmma<<<(tiles + 7) / 8, 256, 0, stream>>>(bufB, Wc, bc, logits, N, D, C);
    }

    // ---- log_softmax ----
    log_softmax_kernel<<<((N * 32) + 255) / 256, 256, 0, stream>>>(logits, lsm, N, C);
}